// STLoss_18391049961934
// MI455X (gfx1250) — compile-verified
//
#include <hip/hip_runtime.h>
#include <math.h>

// ---------------------------------------------------------------------------
// ST-loss pipeline for MI455X (gfx1250, wave32, WMMA).
// Shapes: N_l=N_u=C=128, A=512, N=384.  All fp32 (reference precision).
// Heavy math routed through V_WMMA_F32_16X16X4_F32; batched u^T u uses 64KB LDS.
// ---------------------------------------------------------------------------

typedef float f32x2 __attribute__((ext_vector_type(2)));
typedef float f32x8 __attribute__((ext_vector_type(8)));

#define NL   128
#define NC   128
#define NA   512
#define NTOT 384
#define EPSF 1e-6f
#define ALPHAF 1.2337005501361697f   // pi^2/8
#define RSQRT2F 0.70710678118654752f

// ---- wave32 reductions -----------------------------------------------------
__device__ __forceinline__ float wave_sum(float v) {
#pragma unroll
  for (int m = 16; m > 0; m >>= 1) v += __shfl_xor(v, m, 32);
  return v;
}
__device__ __forceinline__ float wave_max(float v) {
#pragma unroll
  for (int m = 16; m > 0; m >>= 1) v = fmaxf(v, __shfl_xor(v, m, 32));
  return v;
}

// ---- gather feats = [x_l; x_u_w; x_u_s] ------------------------------------
__global__ __launch_bounds__(256) void copy_feats_kernel(
    const float* __restrict__ x_l, const float* __restrict__ x_u_w,
    const float* __restrict__ x_u_s, float* __restrict__ feats) {
  int idx = blockIdx.x * 256 + threadIdx.x;
  if (idx >= NTOT * NA) return;
  int row = idx >> 9;  // /512
  float v;
  if (row < NL)            v = x_l[idx];
  else if (row < 2 * NL)   v = x_u_w[idx - NL * NA];
  else                     v = x_u_s[idx - 2 * NL * NA];
  feats[idx] = v;
}

// ---- per-class amounts + derived scales ------------------------------------
__global__ void amount_kernel(const float* __restrict__ pseudo,
                              float* __restrict__ amount,
                              float* __restrict__ recipAmt,
                              float* __restrict__ qscale) {
  int c = threadIdx.x;  // 128 threads
  float s = 0.f;
  for (int n = 0; n < NL; ++n) s += pseudo[n * NC + c];
  amount[c]   = s;
  recipAmt[c] = (s == 0.f) ? 1.f : 1.f / s;       // amt = where(amount==0,1,amount)
  qscale[c]   = (s > 0.f) ? 1.f / s : 0.f;        // w_cv/amt folded
}

// ---- generic fp32 WMMA GEMM: D[m][n] = sum_k A'[m][k] * B'[k][n] -----------
// TA=0: A is MxK row-major.  TA=1: A is KxM row-major (transposed access).
// TB=1: B source is NxK row-major (i.e. we multiply by B^T, the W case).
// TB=0: B source is KxN row-major.
// One wave computes a 16x16 tile; block = 8 waves = 16 rows x 128 cols.
template <bool TA, bool TB, bool HAS_BIAS, bool HAS_SCALE>
__global__ __launch_bounds__(256) void wmma_gemm_kernel(
    const float* __restrict__ A, const float* __restrict__ B,
    const float* __restrict__ bias, const float* __restrict__ rowScale,
    float* __restrict__ D, int M, int N, int K) {
  const int lane = threadIdx.x & 31;
  const int wave = threadIdx.x >> 5;
  const int m0 = blockIdx.x * 16;
  const int n0 = blockIdx.y * 128 + wave * 16;
  const int li = lane & 15;
  const int kh = (lane >> 4) << 1;  // lanes 16..31 handle K+2,K+3

  f32x8 acc = {0.f, 0.f, 0.f, 0.f, 0.f, 0.f, 0.f, 0.f};

  const float* aP = TA ? (A + (m0 + li))
                       : (A + (size_t)(m0 + li) * K + kh);
  const float* bP = TB ? (B + (size_t)(n0 + li) * K + kh)
                       : (B + (n0 + li));

  for (int k = 0; k < K; k += 4) {
    f32x2 a, b;
    if (TA) {
      int kk = k + kh;
      a[0] = aP[(size_t)kk * M];
      a[1] = aP[(size_t)(kk + 1) * M];
    } else {
      a = *(const f32x2*)(aP + k);
      if ((k & 63) == 0 && k + 64 < K) __builtin_prefetch(aP + k + 64, 0, 0);
    }
    if (TB) {
      b = *(const f32x2*)(bP + k);
      if ((k & 63) == 0 && k + 64 < K) __builtin_prefetch(bP + k + 64, 0, 0);
    } else {
      int kk = k + kh;
      b[0] = bP[(size_t)kk * N];
      b[1] = bP[(size_t)(kk + 1) * N];
    }
    acc = __builtin_amdgcn_wmma_f32_16x16x4_f32(false, a, false, b,
                                                (short)0, acc, false, false);
  }

#pragma unroll
  for (int r = 0; r < 8; ++r) {
    int m = m0 + r + ((lane >> 4) << 3);
    int n = n0 + li;
    float v = acc[r];
    if (HAS_SCALE) v *= rowScale[m];
    if (HAS_BIAS) v += bias[n];
    D[(size_t)m * N + n] = v;
  }
}

// ---- pseudo-label argmaxes: labels[0:128]=pt_l, [128:256]=[256:384]=pt_u ---
__global__ void labels_kernel(const float* __restrict__ y,
                              const float* __restrict__ pseudo,
                              const float* __restrict__ partial_u,
                              int* __restrict__ labels) {
  const int b = blockIdx.x;   // 0..255
  const int lane = threadIdx.x;  // 32
  float bv = -3.4e38f;
  int bi = 0;
#pragma unroll
  for (int s = 0; s < 4; ++s) {
    int j = lane + 32 * s;
    float v;
    if (b < NL) {
      v = pseudo[b * NC + j];
    } else {
      int n = b - NL;
      float yv = y[(NL + n) * NC + j];
      v = yv * fabsf(yv - 1.f) * partial_u[n * NC + j];
    }
    if (v > bv) { bv = v; bi = j; }   // local js increase: '>' keeps first max
  }
#pragma unroll
  for (int m = 16; m > 0; m >>= 1) {
    float ov = __shfl_xor(bv, m, 32);
    int oi = __shfl_xor(bi, m, 32);
    if (ov > bv || (ov == bv && oi < bi)) { bv = ov; bi = oi; }
  }
  if (lane == 0) {
    if (b < NL) labels[b] = bi;
    else { labels[NL + (b - NL)] = bi; labels[2 * NL + (b - NL)] = bi; }
  }
}

// ---- batched Q[c] = qscale[c] * u_c^T u_c, u staged in 64KB LDS ------------
__global__ __launch_bounds__(256) void q_kernel(
    const float* __restrict__ y, const float* __restrict__ bias,
    const float* __restrict__ H, const float* __restrict__ pseudo,
    const float* __restrict__ qscale, float* __restrict__ Q) {
  __shared__ float u[NL * NC];  // 64 KB of the WGP's 320 KB LDS
  const int c = blockIdx.x;
  const int tid = threadIdx.x;

  for (int idx = tid; idx < NL * NC; idx += 256) {
    int n = idx >> 7, i = idx & 127;
    // u[n][i] = lab[n,c] * (G[n,i] - H[c,i]),  G = y_l - b
    u[idx] = pseudo[n * NC + c] * (y[n * NC + i] - bias[i] - H[c * NC + i]);
  }
  __syncthreads();

  const float scale = qscale[c];
  const int lane = tid & 31, wave = tid >> 5;
  const int li = lane & 15, kh = (lane >> 4) << 1;
  float* Qc = Q + (size_t)c * NC * NC;

  for (int t = wave; t < 64; t += 8) {   // 64 16x16 tiles, uniform per wave
    const int i0 = (t >> 3) << 4, j0 = (t & 7) << 4;
    f32x8 acc = {0.f, 0.f, 0.f, 0.f, 0.f, 0.f, 0.f, 0.f};
    for (int k = 0; k < NL; k += 4) {
      int kk = k + kh;
      f32x2 a, b;
      a[0] = u[kk * NC + i0 + li];        // A = u^T : A[i][k] = u[k][i]
      a[1] = u[(kk + 1) * NC + i0 + li];
      b[0] = u[kk * NC + j0 + li];        // B = u   : B[k][j]
      b[1] = u[(kk + 1) * NC + j0 + li];
      acc = __builtin_amdgcn_wmma_f32_16x16x4_f32(false, a, false, b,
                                                  (short)0, acc, false, false);
    }
#pragma unroll
    for (int r = 0; r < 8; ++r) {
      int i = i0 + r + ((lane >> 4) << 3);
      Qc[i * NC + j0 + li] = acc[r] * scale;
    }
  }
}

__global__ void dq_kernel(const float* __restrict__ Q, float* __restrict__ dQ) {
  int c = blockIdx.x, i = threadIdx.x;
  dQ[c * NC + i] = Q[(size_t)c * NC * NC + i * (NC + 1)];
}

// ---- aug[n,i] = clip(softmax_j(y[n,j]+0.5*ratio*quad[c,i,j])[i]) -----------
// quad[c,i,j] = dQ[c,i]+dQ[c,j]-Q[c,i,j]-Q[c,j,i]; diagonal is exactly 0.
__global__ __launch_bounds__(256) void aug_kernel(
    const float* __restrict__ y, const float* __restrict__ Q,
    const float* __restrict__ dQ, const int* __restrict__ labels,
    const float* __restrict__ ratioPtr, float* __restrict__ aug) {
  const int n = blockIdx.y;
  const int i = blockIdx.x * 8 + (threadIdx.x >> 5);
  const int lane = threadIdx.x & 31;
  const float ratio = ratioPtr[0];
  const int c = labels[n];
  const float* yr = y + n * NC;
  const float* dq = dQ + c * NC;
  const float* Qc = Q + (size_t)c * NC * NC;
  const float dqi = dq[i];
  float z[4];
  float lmax = -3.4e38f;
#pragma unroll
  for (int s = 0; s < 4; ++s) {
    int j = lane + 32 * s;
    float quad = dqi + dq[j] - Qc[i * NC + j] - Qc[j * NC + i];
    z[s] = yr[j] + 0.5f * ratio * quad;
    lmax = fmaxf(lmax, z[s]);
  }
  float m = wave_max(lmax);
  float ls = 0.f;
#pragma unroll
  for (int s = 0; s < 4; ++s) ls += expf(z[s] - m);
  float denom = wave_sum(ls);
  if (lane == 0) {
    float v = expf(yr[i] - m) / denom;  // z[i,i] = y[n,i] (quad diag == 0)
    aug[n * NC + i] = fminf(fmaxf(v, EPSF), 1.f - EPSF);
  }
}

// ---- aug1[n,i] = clip(1/(sum_j 1/Phi(...) - C)) over weak-aug rows ---------
__global__ __launch_bounds__(256) void aug1_kernel(
    const float* __restrict__ y, const float* __restrict__ Q,
    const float* __restrict__ dQ, const int* __restrict__ labels,
    const float* __restrict__ ratioPtr, float* __restrict__ aug1) {
  const int n = blockIdx.y;                 // 0..127 (unlabeled)
  const int i = blockIdx.x * 8 + (threadIdx.x >> 5);
  const int lane = threadIdx.x & 31;
  const float ratio = ratioPtr[0];
  const int c = labels[NL + n];             // pt_u[n]
  const float* yr = y + (NL + n) * NC;      // y_u row
  const float* dq = dQ + c * NC;
  const float* Qc = Q + (size_t)c * NC * NC;
  const float dqi = dq[i];
  const float yi = yr[i];
  float ls = 0.f;
#pragma unroll
  for (int s = 0; s < 4; ++s) {
    int j = lane + 32 * s;
    float quad = dqi + dq[j] - Qc[i * NC + j] - Qc[j * NC + i];
    float mu = yi - yr[j];
    float x = ALPHAF * mu / sqrtf(1.f + ALPHAF * ALPHAF * (ratio * quad));
    float sig = 0.5f * (1.f + erff(x * RSQRT2F));
    sig = fminf(fmaxf(sig, EPSF), 1.f - EPSF);
    ls += 1.f / sig;
  }
  float ssum = wave_sum(ls);
  if (lane == 0) {
    float v = 1.f / (ssum - (float)NC);
    aug1[n * NC + i] = fminf(fmaxf(v, EPSF), 1.f - EPSF);
  }
}

// ---- per-row loss pieces (one wave per row; deterministic) -----------------
__global__ void rowloss_labeled_kernel(const float* __restrict__ pseudo,
                                       const float* __restrict__ aug,
                                       float* __restrict__ rl) {
  int n = blockIdx.x, lane = threadIdx.x;
  float acc = 0.f;
#pragma unroll
  for (int s = 0; s < 4; ++s) {
    int i = lane + 32 * s;
    acc += pseudo[n * NC + i] * logf(aug[n * NC + i]);
  }
  float ssum = wave_sum(acc);
  if (lane == 0) rl[n] = -ssum;
}

__global__ void rowloss_unlabeled_kernel(const float* __restrict__ aug1,
                                         const float* __restrict__ partial_u,
                                         const float* __restrict__ aug,
                                         float* __restrict__ rl) {
  int n = blockIdx.x, lane = threadIdx.x;
  const float* augS = aug + (size_t)(2 * NL + n) * NC;  // strong-aug rows
  float a1 = 0.f, a2 = 0.f;
#pragma unroll
  for (int s = 0; s < 4; ++s) {
    int i = lane + 32 * s;
    float w = aug1[n * NC + i] * partial_u[n * NC + i];
    a1 += w;
    a2 += w * logf(augS[i]);
  }
  float s1 = wave_sum(a1);
  float s2 = wave_sum(a2);
  if (lane == 0) rl[n] = -(s2 / s1);   // mask == mask1 == 1 (acc starts at 0)
}

__global__ void rowloss_cl_kernel(const float* __restrict__ partial_l,
                                  const float* __restrict__ partial_u,
                                  const float* __restrict__ aug,
                                  float* __restrict__ rl) {
  int n = blockIdx.x, lane = threadIdx.x;  // n in 0..255
  const float* p = (n < NL) ? (partial_l + n * NC) : (partial_u + (n - NL) * NC);
  float acc = 0.f;
#pragma unroll
  for (int s = 0; s < 4; ++s) {
    int i = lane + 32 * s;
    acc += (1.f - p[i]) * logf(1.0000001f - aug[n * NC + i]);
  }
  float ssum = wave_sum(acc);
  if (lane == 0) rl[n] = -ssum;
}

// ---- deterministic final reduction to the 3 output scalars -----------------
__global__ void finalize_kernel(const float* __restrict__ rl_lab,
                                const float* __restrict__ rl_un,
                                const float* __restrict__ rl_cl,
                                float* __restrict__ out) {
  __shared__ float sh[256];
  int t = threadIdx.x;

  sh[t] = (t < NL) ? rl_lab[t] : 0.f;
  __syncthreads();
  for (int s = 128; s > 0; s >>= 1) { if (t < s) sh[t] += sh[t + s]; __syncthreads(); }
  if (t == 0) out[0] = sh[0] / (float)NL;
  __syncthreads();

  sh[t] = (t < NL) ? rl_un[t] : 0.f;
  __syncthreads();
  for (int s = 128; s > 0; s >>= 1) { if (t < s) sh[t] += sh[t + s]; __syncthreads(); }
  if (t == 0) out[1] = sh[0] / (float)NL;
  __syncthreads();

  sh[t] = rl_cl[t];  // 256 rows
  __syncthreads();
  for (int s = 128; s > 0; s >>= 1) { if (t < s) sh[t] += sh[t + s]; __syncthreads(); }
  if (t == 0) out[2] = sh[0] / 256.f;
}

// ---------------------------------------------------------------------------
// workspace layout (float elements)
// ---------------------------------------------------------------------------
static constexpr size_t OFF_FEATS  = 0;                          // 384*512
static constexpr size_t OFF_Y      = OFF_FEATS  + (size_t)NTOT * NA;
static constexpr size_t OFF_AMOUNT = OFF_Y      + (size_t)NTOT * NC;
static constexpr size_t OFF_RECIP  = OFF_AMOUNT + NC;
static constexpr size_t OFF_QSCALE = OFF_RECIP  + NC;
static constexpr size_t OFF_AVE    = OFF_QSCALE + NC;            // 128*512
static constexpr size_t OFF_H      = OFF_AVE    + (size_t)NC * NA;
static constexpr size_t OFF_Q      = OFF_H      + (size_t)NC * NC;   // 128^3
static constexpr size_t OFF_DQ     = OFF_Q      + (size_t)NC * NC * NC;
static constexpr size_t OFF_AUG    = OFF_DQ     + (size_t)NC * NC;
static constexpr size_t OFF_AUG1   = OFF_AUG    + (size_t)NTOT * NC;
static constexpr size_t OFF_LABELS = OFF_AUG1   + (size_t)NL * NC;   // int[384]
static constexpr size_t OFF_RLLAB  = OFF_LABELS + NTOT;
static constexpr size_t OFF_RLUN   = OFF_RLLAB  + NL;
static constexpr size_t OFF_RLCL   = OFF_RLUN   + NL;            // +256 end

extern "C" void kernel_launch(void* const* d_in, const int* in_sizes, int n_in,
                              void* d_out, int out_size, void* d_ws, size_t ws_size,
                              hipStream_t stream) {
  const float* x_l      = (const float*)d_in[0];
  const float* x_u_w    = (const float*)d_in[1];
  const float* x_u_s    = (const float*)d_in[2];
  const float* W        = (const float*)d_in[3];
  const float* b        = (const float*)d_in[4];
  const float* pseudo_l = (const float*)d_in[5];
  const float* partial_l= (const float*)d_in[6];
  const float* partial_u= (const float*)d_in[7];
  const float* ratio    = (const float*)d_in[8];
  (void)in_sizes; (void)n_in; (void)out_size; (void)ws_size;

  float* ws     = (float*)d_ws;
  float* feats  = ws + OFF_FEATS;
  float* y      = ws + OFF_Y;
  float* amount = ws + OFF_AMOUNT;
  float* recip  = ws + OFF_RECIP;
  float* qscale = ws + OFF_QSCALE;
  float* ave    = ws + OFF_AVE;
  float* H      = ws + OFF_H;
  float* Q      = ws + OFF_Q;
  float* dQ     = ws + OFF_DQ;
  float* aug    = ws + OFF_AUG;
  float* aug1   = ws + OFF_AUG1;
  int*   labels = (int*)(ws + OFF_LABELS);
  float* rl_lab = ws + OFF_RLLAB;
  float* rl_un  = ws + OFF_RLUN;
  float* rl_cl  = ws + OFF_RLCL;
  float* out    = (float*)d_out;

  // 1. feats gather
  copy_feats_kernel<<<(NTOT * NA + 255) / 256, 256, 0, stream>>>(x_l, x_u_w, x_u_s, feats);
  // 2. amounts
  amount_kernel<<<1, 128, 0, stream>>>(pseudo_l, amount, recip, qscale);
  // 3. y = feats @ W^T + b     (M=384, N=128, K=512)
  wmma_gemm_kernel<false, true, true, false><<<dim3(NTOT / 16, 1), 256, 0, stream>>>(
      feats, W, b, nullptr, y, NTOT, NC, NA);
  // 4. ave = (pseudo^T @ x_l) / amt   (M=128, N=512, K=128)
  wmma_gemm_kernel<true, false, false, true><<<dim3(NC / 16, NA / 128), 256, 0, stream>>>(
      pseudo_l, x_l, nullptr, recip, ave, NC, NA, NL);
  // 5. H = ave @ W^T           (M=128, N=128, K=512)
  wmma_gemm_kernel<false, true, false, false><<<dim3(NC / 16, 1), 256, 0, stream>>>(
      ave, W, nullptr, nullptr, H, NC, NC, NA);
  // 6. argmax pseudo-labels
  labels_kernel<<<256, 32, 0, stream>>>(y, pseudo_l, partial_u, labels);
  // 7. Q[c] = qscale[c] * u_c^T u_c  (128 batched 128x128x128 GEMMs in LDS)
  q_kernel<<<NC, 256, 0, stream>>>(y, b, H, pseudo_l, qscale, Q);
  // 8. diag(Q)
  dq_kernel<<<NC, NC, 0, stream>>>(Q, dQ);
  // 9. aug (all 384 rows)
  aug_kernel<<<dim3(16, NTOT), 256, 0, stream>>>(y, Q, dQ, labels, ratio, aug);
  // 10. aug1 (128 weak-aug rows, erf expectation)
  aug1_kernel<<<dim3(16, NL), 256, 0, stream>>>(y, Q, dQ, labels, ratio, aug1);
  // 11. per-row losses
  rowloss_labeled_kernel<<<NL, 32, 0, stream>>>(pseudo_l, aug, rl_lab);
  rowloss_unlabeled_kernel<<<NL, 32, 0, stream>>>(aug1, partial_u, aug, rl_un);
  rowloss_cl_kernel<<<2 * NL, 32, 0, stream>>>(partial_l, partial_u, aug, rl_cl);
  // 12. deterministic scalar reduction -> d_out[0..2]
  finalize_kernel<<<1, 256, 0, stream>>>(rl_lab, rl_un, rl_cl, out);
}